// NgramRF_11158325035418
// MI455X (gfx1250) — compile-verified
//
#include <hip/hip_runtime.h>
#include <hip/hip_bf16.h>
#include <cstdint>

#define N_NODES  100000
#define N_EDGES  1600000
#define N_GRAPHS 4096
#define IN_FEAT  74
#define HIDDEN   128
#define N_CONV   3
#define NGRAM    6
#define BN_EPS   1e-5f

typedef __attribute__((ext_vector_type(2))) float v2f;
typedef __attribute__((ext_vector_type(8))) float v8f;

// -------------------- input projection: h0 = X @ W_in  [N,74]@[74,128] ----
__global__ __launch_bounds__(256) void ngram_inproj(
    const float* __restrict__ X, const float* __restrict__ Win,
    float* __restrict__ h, int n)
{
    int i = blockIdx.x * blockDim.x + threadIdx.x;
    if (i >= n * HIDDEN) return;
    int row = i >> 7;
    int col = i & 127;
    const float* x = X + row * IN_FEAT;
    float acc = 0.f;
    #pragma unroll 2
    for (int k = 0; k < IN_FEAT; ++k)
        acc = fmaf(x[k], Win[k * HIDDEN + col], acc);
    h[i] = acc;
}

// -------------------- softmax over the 6 ngram weights --------------------
__global__ void ngram_softmax6(const float* __restrict__ ngw, float* __restrict__ w)
{
    if (threadIdx.x == 0 && blockIdx.x == 0) {
        float mx = ngw[0];
        for (int i = 1; i < NGRAM; ++i) mx = fmaxf(mx, ngw[i]);
        float e[NGRAM]; float s = 0.f;
        for (int i = 0; i < NGRAM; ++i) { e[i] = __expf(ngw[i] - mx); s += e[i]; }
        float inv = 1.f / s;
        for (int i = 0; i < NGRAM; ++i) w[i] = e[i] * inv;
    }
}

// -------------------- 128x128 weight transpose (done once per conv) -------
__global__ __launch_bounds__(256) void ngram_transpose128(
    const float* __restrict__ W, float* __restrict__ Wt)
{
    int i = blockIdx.x * blockDim.x + threadIdx.x;   // 16384 threads
    int r = i >> 7, c = i & 127;
    Wt[c * HIDDEN + r] = W[i];
}

// -------------------- edge scatter with fused BN+ReLU on the gather -------
// m[dst] += relu(h[src]*scale + shift)   (BN identity when applyBN==0)
// One wave per edge; one float4 per lane; fp32 atomics resolve in L2
// (both 51MB operands fit the 192MB L2).
__global__ __launch_bounds__(256) void ngram_scatter(
    const float* __restrict__ h, float* __restrict__ m,
    const int* __restrict__ src, const int* __restrict__ dst,
    const float* __restrict__ scale, const float* __restrict__ shift,
    int applyBN, int nE)
{
    int gid  = blockIdx.x * blockDim.x + threadIdx.x;
    int e    = gid >> 5;
    if (e >= nE) return;
    int lane = gid & 31;
    int s = src[e];
    int d = dst[e];
    float4 v = *(const float4*)(h + (size_t)s * HIDDEN + lane * 4);
    if (applyBN) {                       // uniform branch across the wave
        const float4 sc = *(const float4*)(scale + lane * 4);
        const float4 sh = *(const float4*)(shift + lane * 4);
        v.x = fmaxf(fmaf(v.x, sc.x, sh.x), 0.f);
        v.y = fmaxf(fmaf(v.y, sc.y, sh.y), 0.f);
        v.z = fmaxf(fmaf(v.z, sc.z, sh.z), 0.f);
        v.w = fmaxf(fmaf(v.w, sc.w, sh.w), 0.f);
    }
    float* mp = m + (size_t)d * HIDDEN + lane * 4;
    unsafeAtomicAdd(mp + 0, v.x);
    unsafeAtomicAdd(mp + 1, v.y);
    unsafeAtomicAdd(mp + 2, v.z);
    unsafeAtomicAdd(mp + 3, v.w);
}

// -------------------- conv GEMM: Y = M @ W  (exact fp32 WMMA) -------------
// Block = 256 threads = 8 waves; block owns 16 rows x 128 cols.
// Wave w computes the 16x16 tile at columns [16w,16w+16) with
// V_WMMA_F32_16X16X4_F32 over K=128 (32 MACs). BN sum/sumsq fused.
// A staged in LDS (stride 130 -> bank-conflict-free AND 8B-aligned b64
// loads); B read as contiguous b64 pairs from the transposed weights.
__global__ __launch_bounds__(256) void ngram_conv_gemm_wmma(
    const float* __restrict__ M, const float* __restrict__ Wt,
    float* __restrict__ Y, float* __restrict__ colsum,
    float* __restrict__ colsumsq, int n)
{
    __shared__ float ldsA[16 * 130];

    const int rbase = blockIdx.x * 16;
    const int tid   = threadIdx.x;

    // Stage the 16x128 row tile of M (coalesced), shared by all 8 waves.
    for (int i = tid; i < 16 * HIDDEN; i += 256) {
        int r = i >> 7, k = i & 127;
        ldsA[r * 130 + k] = M[(size_t)(rbase + r) * HIDDEN + k];
    }
    __syncthreads();

    const int wave  = tid >> 5;        // 0..7 -> column tile
    const int lane  = tid & 31;
    const int half  = lane >> 4;       // 0: lanes 0-15, 1: lanes 16-31
    const int lr    = lane & 15;
    const int cbase = wave * 16;

    const float* ldA = &ldsA[lr * 130];                 // row lr of A tile
    const float* wB  = &Wt[(size_t)(cbase + lr) * HIDDEN]; // column (cbase+lr) of W

    v8f c = {};                        // fp32 16x16 accumulator (8 VGPRs)
    for (int kt = 0; kt < HIDDEN; kt += 4) {
        const int k0 = kt + half * 2;  // even -> 8B aligned
        v2f a = *(const v2f*)(ldA + k0);   // A: row lr, K = k0,k0+1
        v2f b = *(const v2f*)(wB  + k0);   // B: col lr, K = k0,k0+1
        c = __builtin_amdgcn_wmma_f32_16x16x4_f32(
                /*neg_a=*/false, a, /*neg_b=*/false, b,
                /*c_mod=*/(short)0, c, /*reuse_a=*/false, /*reuse_b=*/false);
    }

    // D layout: VGPR v -> row (v + 8*half), col lr. Store + BN partial stats.
    const int col = cbase + lr;
    float s = 0.f, s2 = 0.f;
    #pragma unroll
    for (int v = 0; v < 8; ++v) {
        float val = c[v];
        int   row = rbase + v + half * 8;
        Y[(size_t)row * HIDDEN + col] = val;
        s  += val;
        s2 += val * val;
    }
    // combine the two half-waves holding the same column
    s  += __shfl_xor(s, 16, 32);
    s2 += __shfl_xor(s2, 16, 32);
    if (half == 0) {
        unsafeAtomicAdd(&colsum[col],   s);
        unsafeAtomicAdd(&colsumsq[col], s2);
    }
}

// -------------------- BN stat finalize (1 block, 128 threads) -------------
__global__ void ngram_bn_finalize(
    const float* __restrict__ colsum, const float* __restrict__ colsumsq,
    const float* __restrict__ gamma, const float* __restrict__ beta,
    float* __restrict__ scale, float* __restrict__ shift, int n)
{
    int c = threadIdx.x;
    if (c >= HIDDEN) return;
    float invn = 1.f / (float)n;
    float mean = colsum[c] * invn;
    float var  = fmaxf(colsumsq[c] * invn - mean * mean, 0.f);
    float sc   = gamma[c] * rsqrtf(var + BN_EPS);
    scale[c] = sc;
    shift[c] = beta[c] - mean * sc;
}

// -------------------- weighted pooling with fused BN+ReLU -----------------
// combined[g] += w[it] * relu(y*scale + shift)
__global__ __launch_bounds__(256) void ngram_pool(
    const float* __restrict__ y, const int* __restrict__ gids,
    const float* __restrict__ wsm, int it,
    const float* __restrict__ scale, const float* __restrict__ shift,
    float* __restrict__ combined, int n)
{
    int i = blockIdx.x * blockDim.x + threadIdx.x;
    if (i >= n * HIDDEN) return;
    int node = i >> 7;
    int col  = i & 127;
    float v  = fmaxf(fmaf(y[i], scale[col], shift[col]), 0.f);
    unsafeAtomicAdd(&combined[(size_t)gids[node] * HIDDEN + col], wsm[it] * v);
}

// -------------------- MLP head: leaky_relu(X@W1+b1) @ W2 + b2 -> sigmoid --
__global__ __launch_bounds__(128) void ngram_mlp_head(
    const float* __restrict__ combined, const float* __restrict__ W1,
    const float* __restrict__ b1, const float* __restrict__ W2,
    const float* __restrict__ b2, float* __restrict__ out)
{
    __shared__ float red[HIDDEN];
    int g = blockIdx.x;
    int t = threadIdx.x;
    const float* row = combined + (size_t)g * HIDDEN;
    float acc = b1[t];
    #pragma unroll 4
    for (int k = 0; k < HIDDEN; ++k)
        acc = fmaf(row[k], W1[k * HIDDEN + t], acc);
    acc = acc > 0.f ? acc : 0.01f * acc;       // leaky relu
    red[t] = acc * W2[t];                      // OUT == 1
    __syncthreads();
    for (int off = 64; off > 0; off >>= 1) {
        if (t < off) red[t] += red[t + off];
        __syncthreads();
    }
    if (t == 0)
        out[g] = 1.f / (1.f + __expf(-(red[0] + b2[0])));
}

// ==========================================================================
extern "C" void kernel_launch(void* const* d_in, const int* in_sizes, int n_in,
                              void* d_out, int out_size, void* d_ws, size_t ws_size,
                              hipStream_t stream)
{
    const float* features = (const float*)d_in[0];
    const float* W_in     = (const float*)d_in[1];
    const float* conv_w   = (const float*)d_in[2];
    const float* bn_gamma = (const float*)d_in[3];
    const float* bn_beta  = (const float*)d_in[4];
    const float* ngram_w  = (const float*)d_in[5];
    const float* W1       = (const float*)d_in[6];
    const float* b1       = (const float*)d_in[7];
    const float* W2       = (const float*)d_in[8];
    const float* b2       = (const float*)d_in[9];
    const int*   src      = (const int*)d_in[10];
    const int*   dst      = (const int*)d_in[11];
    const int*   gids     = (const int*)d_in[12];
    float*       out      = (float*)d_out;

    const size_t NH = (size_t)N_NODES * HIDDEN;
    float* h0       = (float*)d_ws;            // [N,128] persistent input proj
    float* bufA     = h0   + NH;               // rotating: aggregation target
    float* bufB     = bufA + NH;               // rotating: GEMM output (pre-BN)
    float* combined = bufB + NH;               // [4096,128]
    float* colsum   = combined + (size_t)N_GRAPHS * HIDDEN;   // [128]
    float* colsumsq = colsum + HIDDEN;                         // [128]
    float* bnscale  = colsumsq + HIDDEN;                       // [128]
    float* bnshift  = bnscale + HIDDEN;                        // [128]
    float* wsm      = bnshift + HIDDEN;                        // [6] (+pad)
    float* Wt       = wsm + 32;                // [3,128,128] transposed weights

    const int elemN = N_NODES * HIDDEN;        // 12.8M
    const int gElem = (elemN + 255) / 256;
    const int gScat = (N_EDGES * 32 + 255) / 256;

    ngram_inproj<<<gElem, 256, 0, stream>>>(features, W_in, h0, N_NODES);
    ngram_softmax6<<<1, 32, 0, stream>>>(ngram_w, wsm);
    for (int j = 0; j < N_CONV; ++j)
        ngram_transpose128<<<HIDDEN * HIDDEN / 256, 256, 0, stream>>>(
            conv_w + (size_t)j * HIDDEN * HIDDEN, Wt + (size_t)j * HIDDEN * HIDDEN);
    hipMemsetAsync(combined, 0, (size_t)N_GRAPHS * HIDDEN * sizeof(float), stream);

    for (int i = 0; i < NGRAM; ++i) {
        const float* cur = h0;                 // every ngram pass restarts from h0
        for (int j = 0; j < N_CONV; ++j) {
            hipMemsetAsync(bufA, 0, NH * sizeof(float), stream);
            // gather applies BN+ReLU of the *previous* conv (identity for j==0,
            // where cur == h0 which is already final)
            ngram_scatter<<<gScat, 256, 0, stream>>>(
                cur, bufA, src, dst, bnscale, bnshift, (j > 0) ? 1 : 0, N_EDGES);
            hipMemsetAsync(colsum, 0, 2 * HIDDEN * sizeof(float), stream);
            ngram_conv_gemm_wmma<<<N_NODES / 16, 256, 0, stream>>>(
                bufA, Wt + (size_t)j * HIDDEN * HIDDEN, bufB,
                colsum, colsumsq, N_NODES);
            ngram_bn_finalize<<<1, 128, 0, stream>>>(
                colsum, colsumsq, bn_gamma + j * HIDDEN, bn_beta + j * HIDDEN,
                bnscale, bnshift, N_NODES);
            cur = bufB;                        // pre-BN; consumers fuse BN+ReLU
        }
        // pool conv-2 output with its BN+ReLU fused, weighted by softmax w[i]
        ngram_pool<<<gElem, 256, 0, stream>>>(
            bufB, gids, wsm, i, bnscale, bnshift, combined, N_NODES);
    }

    ngram_mlp_head<<<N_GRAPHS, 128, 0, stream>>>(combined, W1, b1, W2, b2, out);
}